// ProsodyExtractor_13494787244693
// MI455X (gfx1250) — compile-verified
//
#include <hip/hip_runtime.h>
#include <math.h>

typedef float v2f __attribute__((ext_vector_type(2)));
typedef float v8f __attribute__((ext_vector_type(8)));

constexpr int BB = 8;      // batch
constexpr int SS = 100;    // segments
constexpr int NM = 80;     // n_mel
constexpr int DMH = 128;   // hidden
constexpr int KK = 9;      // conv kernel
constexpr int TT = 1200;   // S * DMAX
constexpr int GG = 384;    // 3*DM
constexpr float EPSV = 1e-5f;

__device__ __forceinline__ float fast_sigmoid(float x) {
    return __builtin_amdgcn_rcpf(1.f + __expf(-x));
}
__device__ __forceinline__ float fast_tanh(float x) {
    // tanh(x) = 1 - 2/(exp(2x)+1); saturates correctly at +/-inf
    return 1.f - 2.f * __builtin_amdgcn_rcpf(__expf(2.f * x) + 1.f);
}

// ---------------------------------------------------------------------------
// Kernel 0: per-batch prefix sums -> starts/ends + reset masks
// ---------------------------------------------------------------------------
__global__ void prep_kernel(const int* __restrict__ durations,
                            float* __restrict__ startmask, float* __restrict__ endmask,
                            int* __restrict__ starts, int* __restrict__ ends) {
    int b = blockIdx.x;
    int lane = threadIdx.x;
    for (int i = lane; i < TT; i += 32) { startmask[b*TT + i] = 0.f; endmask[b*TT + i] = 0.f; }
    __syncthreads();
    if (lane == 0) {
        int cum = 0;
        for (int s = 0; s < SS; ++s) {
            int d = durations[b*SS + s];
            int st = cum; cum += d; int en = cum - 1;
            starts[b*SS + s] = st;
            ends[b*SS + s]   = en;
            startmask[b*TT + st] = 1.f;
            endmask[b*TT + en]   = 1.f;
        }
    }
}

// ---------------------------------------------------------------------------
// Kernel 1: conv(1->128, K=9 over mel bins) + BN + ReLU + conv(128->1) + BN + ReLU
// One block per (b,t) row. LDS holds the padded mel row and per-channel row.
// ---------------------------------------------------------------------------
__global__ void conv_kernel(const float* __restrict__ mel,
                            const float* __restrict__ w1, const float* __restrict__ g1,
                            const float* __restrict__ be1,
                            const float* __restrict__ w2, const float* __restrict__ g2,
                            const float* __restrict__ be2,
                            float* __restrict__ xout) {
    __shared__ __align__(16) float smel[NM + 8];
    __shared__ __align__(16) float sv[NM + 8];
    __shared__ float sw1[DMH * KK];
    __shared__ float sw2[DMH * KK];
    __shared__ float ss1[DMH], sb1[DMH];
    const int row = blockIdx.x;          // b*TT + t
    const int tid = threadIdx.x;         // 0..95
    const float inv = rsqrtf(1.0f + EPSV);
    for (int i = tid; i < DMH * KK; i += blockDim.x) { sw1[i] = w1[i]; sw2[i] = w2[i]; }
    for (int i = tid; i < DMH; i += blockDim.x)      { ss1[i] = g1[i] * inv; sb1[i] = be1[i]; }
    if (tid < 4) { smel[tid] = 0.f; smel[NM + 4 + tid] = 0.f; sv[tid] = 0.f; sv[NM + 4 + tid] = 0.f; }
    if (tid < NM) smel[4 + tid] = mel[row * NM + tid];
    __syncthreads();
    float acc = 0.f;
    for (int c = 0; c < DMH; ++c) {
        if (tid < NM) {
            float c1 = 0.f;
            #pragma unroll
            for (int k = 0; k < KK; ++k) c1 += sw1[c*KK + k] * smel[tid + k];
            sv[4 + tid] = fmaxf(c1 * ss1[c] + sb1[c], 0.f);
        }
        __syncthreads();
        if (tid < NM) {
            #pragma unroll
            for (int k = 0; k < KK; ++k) acc += sw2[c*KK + k] * sv[tid + k];
        }
        __syncthreads();
    }
    if (tid < NM) {
        xout[row * NM + tid] = fmaxf(acc * (g2[0] * inv) + be2[0], 0.f);
    }
}

// ---------------------------------------------------------------------------
// Kernel 2: input projections via f32 WMMA.  C = x(9600x80) @ w_ihT(80x384) + b
// One 16x16 output tile per wave, K=80 -> 20 chained v_wmma_f32_16x16x4_f32.
// Output layout xg[t][b][g] (scan-friendly).
// ---------------------------------------------------------------------------
__global__ void proj_kernel(const float* __restrict__ x,
                            const float* __restrict__ w_ih_f, const float* __restrict__ b_ih_f,
                            const float* __restrict__ w_ih_b, const float* __restrict__ b_ih_b,
                            float* __restrict__ xf, float* __restrict__ xb) {
    const int lane  = threadIdx.x & 31;
    const int wave  = threadIdx.x >> 5;
    const int tile  = blockIdx.x * 8 + wave;       // 600*24*2 tiles total
    const int dir   = tile / (600 * 24);
    const int rem   = tile % (600 * 24);
    const int mt    = rem / 24;
    const int nt    = rem % 24;
    const float* W    = dir ? w_ih_b : w_ih_f;     // (384,80) row-major
    const float* bias = dir ? b_ih_b : b_ih_f;
    float*       out  = dir ? xb : xf;             // (TT,BB,GG)

    const int rlo   = lane & 15;
    const int khalf = lane >> 4;
    const int m = mt * 16 + rlo;
    const int n = nt * 16 + rlo;

    v8f acc = {0.f, 0.f, 0.f, 0.f, 0.f, 0.f, 0.f, 0.f};
    #pragma unroll
    for (int ks = 0; ks < 20; ++ks) {
        const int k = ks * 4 + 2 * khalf;
        v2f a = *(const v2f*)(x + (size_t)m * NM + k);
        v2f b = *(const v2f*)(W + (size_t)n * NM + k);
        acc = __builtin_amdgcn_wmma_f32_16x16x4_f32(false, a, false, b, (short)0, acc, false, false);
    }
    const float bv = bias[nt * 16 + rlo];
    #pragma unroll
    for (int i = 0; i < 8; ++i) {
        const int mrow = mt * 16 + i + 8 * khalf;
        const int bidx = mrow / TT;
        const int t    = mrow % TT;
        out[((size_t)t * BB + bidx) * GG + nt * 16 + rlo] = acc[i] + bv;
    }
}

// ---------------------------------------------------------------------------
// Kernel 3: bidirectional GRU scan.  grid=2 (dir), block=768 (24 waves).
// Each wave owns one 16-wide slice of the 384 gate outputs and pre-loads its
// w_hh B-fragments (64 VGPRs) once; per step: 32 chained WMMAs on h from LDS.
// Gate inputs (xg) are loaded at the TOP of each step so the global-memory
// latency overlaps the WMMA chain; next step's xg row is prefetched.
// ---------------------------------------------------------------------------
__global__ void gru_kernel(const float* __restrict__ xf, const float* __restrict__ xb,
                           const float* __restrict__ w_hh_f, const float* __restrict__ b_hh_f,
                           const float* __restrict__ w_hh_b, const float* __restrict__ b_hh_b,
                           const float* __restrict__ startmask, const float* __restrict__ endmask,
                           float* __restrict__ Hf, float* __restrict__ Hb) {
    const int dir = blockIdx.x;
    const float* xg   = dir ? xb : xf;
    const float* W    = dir ? w_hh_b : w_hh_f;     // (384,128) row-major
    const float* bh   = dir ? b_hh_b : b_hh_f;
    const float* mask = dir ? endmask : startmask;
    float*       H    = dir ? Hb : Hf;

    __shared__ __align__(16) float hL[16 * DMH];   // rows 8..15 stay zero
    __shared__ __align__(16) float hgL[BB * GG];
    __shared__ float bhL[GG];

    const int tid   = threadIdx.x;                 // 0..767
    const int lane  = tid & 31;
    const int wave  = tid >> 5;                    // 0..23 -> N tile
    const int rlo   = lane & 15;
    const int khalf = lane >> 4;
    const int n0    = wave * 16;
    const int ncol  = n0 + rlo;

    for (int i = tid; i < 16 * DMH; i += 768) hL[i] = 0.f;
    for (int i = tid; i < GG; i += 768)       bhL[i] = bh[i];

    // Element ownership for reset + gate phases (same mapping both phases)
    const int e0 = tid;                // always < 1024
    const int e1 = tid + 768;          // valid only for tid < 256
    const int b0 = e0 >> 7, j0 = e0 & 127;
    const int b1 = e1 >> 7, j1 = e1 & 127;
    const bool has1 = (e1 < BB * DMH);

    // Pre-load recurrent weight fragments: B[k][n] = w_hh[n][k]
    v2f bf[32];
    #pragma unroll
    for (int ks = 0; ks < 32; ++ks)
        bf[ks] = *(const v2f*)(W + (size_t)ncol * DMH + ks * 4 + 2 * khalf);
    __syncthreads();

    for (int s = 0; s < TT; ++s) {
        const int ta = dir ? (TT - 1 - s) : s;
        const float* xrow = xg + (size_t)ta * BB * GG;

        // --- issue this step's independent global loads early -------------
        const float xr0 = xrow[b0*GG + j0];
        const float xz0 = xrow[b0*GG + 128 + j0];
        const float xn0 = xrow[b0*GG + 256 + j0];
        float xr1 = 0.f, xz1 = 0.f, xn1 = 0.f;
        if (has1) {
            xr1 = xrow[b1*GG + j1];
            xz1 = xrow[b1*GG + 128 + j1];
            xn1 = xrow[b1*GG + 256 + j1];
        }
        const float rst0 = mask[b0*TT + ta];
        const float rst1 = has1 ? mask[b1*TT + ta] : 0.f;

        // --- h *= (1 - reset) ---------------------------------------------
        hL[e0] *= (1.f - rst0);
        if (has1) hL[e1] *= (1.f - rst1);
        __syncthreads();

        // --- hg tile = h @ w_hhT  (32 chained f32 WMMAs) ------------------
        v8f acc = {0.f, 0.f, 0.f, 0.f, 0.f, 0.f, 0.f, 0.f};
        const float* arow = hL + rlo * DMH;
        #pragma unroll
        for (int ks = 0; ks < 32; ++ks) {
            v2f a = *(const v2f*)(arow + ks * 4 + 2 * khalf);
            acc = __builtin_amdgcn_wmma_f32_16x16x4_f32(false, a, false, bf[ks], (short)0, acc, false, false);
        }
        if (lane < 16) {            // rows M=0..7 (valid batch rows) live in lanes 0-15
            #pragma unroll
            for (int i = 0; i < 8; ++i) hgL[i * GG + n0 + lane] = acc[i];
        }

        // --- prefetch next step's xg row (global_prefetch_b8) -------------
        if (s + 1 < TT) {
            const int tn = dir ? (ta - 1) : (ta + 1);
            __builtin_prefetch(xg + (size_t)tn * BB * GG + tid * 4, 0, 1);
        }
        __syncthreads();

        // --- gates ---------------------------------------------------------
        {
            const float hr = hgL[b0*GG + j0]       + bhL[j0];
            const float hz = hgL[b0*GG + 128 + j0] + bhL[128 + j0];
            const float hn = hgL[b0*GG + 256 + j0] + bhL[256 + j0];
            const float r = fast_sigmoid(xr0 + hr);
            const float z = fast_sigmoid(xz0 + hz);
            const float n = fast_tanh(xn0 + r * hn);
            const float hnew = (1.f - z) * n + z * hL[e0];
            hL[e0] = hnew;
            H[(size_t)ta * BB * DMH + e0] = hnew;
        }
        if (has1) {
            const float hr = hgL[b1*GG + j1]       + bhL[j1];
            const float hz = hgL[b1*GG + 128 + j1] + bhL[128 + j1];
            const float hn = hgL[b1*GG + 256 + j1] + bhL[256 + j1];
            const float r = fast_sigmoid(xr1 + hr);
            const float z = fast_sigmoid(xz1 + hz);
            const float n = fast_tanh(xn1 + r * hn);
            const float hnew = (1.f - z) * n + z * hL[e1];
            hL[e1] = hnew;
            H[(size_t)ta * BB * DMH + e1] = hnew;
        }
        __syncthreads();
    }
}

// ---------------------------------------------------------------------------
// Kernel 4: gather hidden states at segment ends (fwd) / starts (bwd), mask.
// ---------------------------------------------------------------------------
__global__ void gather_kernel(const float* __restrict__ Hf, const float* __restrict__ Hb,
                              const int* __restrict__ starts, const int* __restrict__ ends,
                              const int* __restrict__ src_len, float* __restrict__ out) {
    const int bs = blockIdx.x;        // b*SS + s
    const int b = bs / SS, s = bs % SS;
    const int tid = threadIdx.x;      // 0..255
    const bool valid = s < src_len[b];
    float val;
    if (tid < DMH) {
        const int e = ends[bs];
        val = Hf[((size_t)e * BB + b) * DMH + tid];
    } else {
        const int st = starts[bs];
        val = Hb[((size_t)st * BB + b) * DMH + (tid - DMH)];
    }
    out[(size_t)bs * 256 + tid] = valid ? val : 0.f;
}

// ---------------------------------------------------------------------------
extern "C" void kernel_launch(void* const* d_in, const int* in_sizes, int n_in,
                              void* d_out, int out_size, void* d_ws, size_t ws_size,
                              hipStream_t stream) {
    const float* mel       = (const float*)d_in[0];
    const int*   durations = (const int*)  d_in[1];
    // d_in[2] = mel_len (unused)
    const int*   src_len   = (const int*)  d_in[3];
    const float* w1  = (const float*)d_in[4];
    const float* g1  = (const float*)d_in[5];
    const float* be1 = (const float*)d_in[6];
    const float* w2  = (const float*)d_in[7];
    const float* g2  = (const float*)d_in[8];
    const float* be2 = (const float*)d_in[9];
    const float* w_ih_f = (const float*)d_in[10];
    const float* w_hh_f = (const float*)d_in[11];
    const float* b_ih_f = (const float*)d_in[12];
    const float* b_hh_f = (const float*)d_in[13];
    const float* w_ih_b = (const float*)d_in[14];
    const float* w_hh_b = (const float*)d_in[15];
    const float* b_ih_b = (const float*)d_in[16];
    const float* b_hh_b = (const float*)d_in[17];

    char* ws = (char*)d_ws;
    size_t off = 0;
    auto take = [&](size_t bytes) -> void* {
        void* p = ws + off;
        off = (off + bytes + 255) & ~(size_t)255;
        return p;
    };
    float* x         = (float*)take((size_t)BB * TT * NM * 4);     // 3.0 MB
    float* xf        = (float*)take((size_t)TT * BB * GG * 4);     // 14.7 MB
    float* xb        = (float*)take((size_t)TT * BB * GG * 4);     // 14.7 MB
    float* Hf        = (float*)take((size_t)TT * BB * DMH * 4);    // 4.9 MB
    float* Hb        = (float*)take((size_t)TT * BB * DMH * 4);    // 4.9 MB
    float* startmask = (float*)take((size_t)BB * TT * 4);
    float* endmask   = (float*)take((size_t)BB * TT * 4);
    int*   starts    = (int*)  take((size_t)BB * SS * 4);
    int*   ends      = (int*)  take((size_t)BB * SS * 4);
    (void)ws_size; (void)in_sizes; (void)n_in; (void)out_size;

    prep_kernel<<<BB, 32, 0, stream>>>(durations, startmask, endmask, starts, ends);
    conv_kernel<<<BB * TT, 96, 0, stream>>>(mel, w1, g1, be1, w2, g2, be2, x);
    proj_kernel<<<(600 * 24 * 2) / 8, 256, 0, stream>>>(x, w_ih_f, b_ih_f, w_ih_b, b_ih_b, xf, xb);
    gru_kernel<<<2, 768, 0, stream>>>(xf, xb, w_hh_f, b_hh_f, w_hh_b, b_hh_b,
                                      startmask, endmask, Hf, Hb);
    gather_kernel<<<BB * SS, 256, 0, stream>>>(Hf, Hb, starts, ends, src_len, (float*)d_out);
}